// RoiAlign_43542378447408
// MI455X (gfx1250) — compile-verified
//
#include <hip/hip_runtime.h>
#include <hip/hip_bf16.h>

// ROI-Align (POOL=7) for MI455X / gfx1250.
//
// Perf model: 64 MB feature map is fully L2-resident (192 MB L2); the kernel is
// a latency-bound gather (4 x b32 per cell) -> maximize wave32 occupancy, keep
// VGPR use tiny, stream the output with non-temporal stores so it never evicts
// the feature map from L2. ROI descriptors are staged into LDS once per block
// via the CDNA5 async global->LDS DMA path (ASYNCcnt) instead of 49x-redundant
// per-thread global loads. No matrix structure exists in this op, so WMMA is
// not applicable; the CDNA5-specific instructions used are the async-to-LDS
// ops + s_wait_asynccnt.

#define POOL 7
#define CELLS (POOL * POOL)        // 49 cells per ROI
#define ROIS_PER_BLOCK 5           // 5 * 49 = 245 worker threads
#define BLOCK_THREADS 256          // 8 wave32 waves
#define W_SHIFT 10                 // W = 1024
#define B_SHIFT 20                 // H*W = 1<<20

#if defined(__AMDGCN__) && __has_builtin(__builtin_amdgcn_global_load_async_to_lds_b32)
#define USE_ASYNC_LDS 1
#else
#define USE_ASYNC_LDS 0
#endif

#if USE_ASYNC_LDS
typedef __attribute__((address_space(1))) int* gint_ptr;   // global dword
typedef __attribute__((address_space(3))) int* lint_ptr;   // LDS dword
#endif

__global__ __launch_bounds__(BLOCK_THREADS)
void roi_align_pool7_kernel(const float* __restrict__ fm,
                            const float* __restrict__ rois,
                            float* __restrict__ out,
                            int N)
{
    __shared__ float sroi[ROIS_PER_BLOCK * 5];

    const int tid     = threadIdx.x;
    const int roiBase = blockIdx.x * ROIS_PER_BLOCK;
    // floats of ROI descriptor data this block must stage (last block may be short)
    const int nvals   = min(ROIS_PER_BLOCK * 5, N * 5 - roiBase * 5);

    if (tid < nvals) {
#if USE_ASYNC_LDS
        // CDNA5 async global -> LDS copy, one dword per lane (tracked by ASYNCcnt).
        __builtin_amdgcn_global_load_async_to_lds_b32(
            (gint_ptr)(rois + roiBase * 5 + tid),
            (lint_ptr)(&sroi[tid]),
            /*imm offset*/ 0, /*cpol*/ 0);
#else
        sroi[tid] = rois[roiBase * 5 + tid];
#endif
    }
#if USE_ASYNC_LDS
#if __has_builtin(__builtin_amdgcn_s_wait_asynccnt)
    __builtin_amdgcn_s_wait_asynccnt(0);
#else
    asm volatile("s_wait_asynccnt 0x0" ::: "memory");
#endif
#endif
    __syncthreads();

    if (tid >= ROIS_PER_BLOCK * CELLS) return;   // 11 tail threads idle

    const int nLoc = tid / CELLS;                // ROI within block
    const int c    = tid - nLoc * CELLS;         // cell 0..48
    const int n    = roiBase + nLoc;
    if (n >= N) return;

    const int i = c / POOL;                      // y bin
    const int j = c - i * POOL;                  // x bin

    const float* r  = &sroi[nLoc * 5];
    const int    b  = (int)r[0];
    const float  x1 = r[1];
    const float  y1 = r[2];
    const float  rw = r[3];
    const float  rh = r[4];

    // t_k = k * (1/7), matching jnp.linspace(0, 1, 8)
    const float inv7 = 1.0f / 7.0f;
    const float gx0  = x1 + rw * ((float)j       * inv7);
    const float gx1  = x1 + rw * ((float)(j + 1) * inv7);
    const float gy0  = y1 + rh * ((float)i       * inv7);
    const float gy1  = y1 + rh * ((float)(i + 1) * inv7);

    const float xl_f = floorf(gx0);
    const float xr_f = ceilf(gx1);
    const float yt_f = floorf(gy0);
    const float yb_f = ceilf(gy1);

    const float x_l = gx0 - xl_f;
    const float x_r = xr_f - gx0;
    const float y_t = gy0 - yt_f;
    const float y_b = yb_f - gy0;

    const int xl = (int)xl_f;
    const int yt = (int)yt_f;

    // fm index = ((b*1024) + yt)*1024 + xl ; reference guarantees xl+1 <= W-1,
    // yt+1 <= H-1, so no clamping (same as the reference).
    const float* p = fm + (((size_t)b << B_SHIFT) + ((size_t)yt << W_SHIFT) + (size_t)xl);

    const float f00 = p[0];
    const float f01 = p[1];
    const float f10 = p[1 << W_SHIFT];
    const float f11 = p[(1 << W_SHIFT) + 1];

    const float num = f00 * x_r * y_b + f01 * x_l * y_b
                    + f10 * x_r * y_t + f11 * x_l * y_t;
    const float den = (x_l + x_r) * (y_t + y_b);
    const float v   = num / den;

    // Output is write-once & streaming: non-temporal store keeps the 64 MB
    // feature map resident in the 192 MB L2 (the key working set here).
    __builtin_nontemporal_store(v, &out[(size_t)n * CELLS + (size_t)c]);
}

extern "C" void kernel_launch(void* const* d_in, const int* in_sizes, int n_in,
                              void* d_out, int out_size, void* d_ws, size_t ws_size,
                              hipStream_t stream) {
    const float* fm   = (const float*)d_in[0];   // (16, 1024, 1024) f32
    const float* rois = (const float*)d_in[1];   // (N, 5) f32
    float* out        = (float*)d_out;           // (N, 7, 7) f32

    const int N      = in_sizes[1] / 5;
    const int blocks = (N + ROIS_PER_BLOCK - 1) / ROIS_PER_BLOCK;

    roi_align_pool7_kernel<<<blocks, BLOCK_THREADS, 0, stream>>>(fm, rois, out, N);
}